// GroupQuantization_68934225101057
// MI455X (gfx1250) — compile-verified
//
#include <hip/hip_runtime.h>

typedef __attribute__((ext_vector_type(2))) float v2f;
typedef __attribute__((ext_vector_type(8))) float v8f;

#define GR 6
#define BS 64
#define HH 12
#define WW 1024
#define CC 64
#define LL 256            // WW / NUM_OVERLAPS
#define MM 16384          // BS * LL
#define KK 512
#define NN 1024
#define KN (KK*NN)        // 524288
#define DECAYF 0.99f
#define OMD 0.01f
#define EPSF 1e-5f

#define MROWS 32          // M rows per block (2 x 16-row WMMA sub-tiles)
#define LDA 514           // padded LDS stride (floats) for A tile

// ---- workspace layout (float offsets) ----
#define WS_ACC    0                         // GR*NN*KK  acc[g][n][k]  (transposed emb_sum)
#define WS_COUNTS (WS_ACC + GR*KN)          // GR*NN
#define WS_LOSS   (WS_COUNTS + GR*NN)       // 1 (+7 pad)
#define WS_NSUM   (WS_LOSS + 8)             // GR (+2 pad)
#define WS_E2     (WS_NSUM + 8)             // GR*NN
#define WS_ET     (WS_E2 + GR*NN)           // GR*NN*KK  Et[g][n][k]
#define WS_CODES  (WS_ET + GR*KN)           // GR*MM ints
#define WS_ZERO_N WS_E2                     // region [0, WS_E2) zeroed each call

// ---- output layout (float offsets) ----
#define XQ_SZ    (BS*HH*WW*CC)              // 50331648
#define OFF_LOSS XQ_SZ
#define OFF_EMB  (OFF_LOSS + 1)
#define OFF_CS   (OFF_EMB + GR*KN)

// zg[g, m, k] <-> z[b, h*W + w, c] mapping (pure reshape/transpose bijection)
__device__ __forceinline__ long z_index(int b, int l, int g, int k) {
  int idx = g * KK + k;          // 0..3071 within (o,h,c)
  int o = idx / 768;
  int rem = idx - o * 768;
  int h = rem >> 6;
  int c = rem & 63;
  int w = l * 4 + o;
  return (long)b * (HH * WW * CC) + (long)h * (WW * CC) + (long)w * CC + c;
}

// ---------------- init: zero accumulators ----------------
__global__ void k_init(float* __restrict__ ws) {
  int t = blockIdx.x * blockDim.x + threadIdx.x;
  if (t < WS_ZERO_N) ws[t] = 0.0f;
}

// ---------------- e2[g][n] = sum_k emb[g][k][n]^2 ----------------
__global__ void k_e2(const float* __restrict__ emb, float* __restrict__ e2) {
  int t = blockIdx.x * blockDim.x + threadIdx.x;   // GR*NN threads
  int g = t / NN;
  int n = t - g * NN;
  const float* p = emb + (long)g * KN + n;
  float s = 0.0f;
  for (int k = 0; k < KK; ++k) s += p[(long)k * NN] * p[(long)k * NN];
  e2[t] = s;
}

// ---------------- Et[g][n][k] = emb[g][k][n] (tiled transpose) ----------------
__global__ void k_transpose(const float* __restrict__ emb, float* __restrict__ Et) {
  __shared__ float tile[32][33];
  int g = blockIdx.z;
  int k0 = blockIdx.x * 32;
  int n0 = blockIdx.y * 32;
  int col = threadIdx.x & 31;
  int row0 = threadIdx.x >> 5;     // 0..7
  const float* src = emb + (long)g * KN;
  float* dst = Et + (long)g * KN;
#pragma unroll
  for (int p = 0; p < 4; ++p) {
    int r = row0 + p * 8;
    tile[r][col] = src[(long)(k0 + r) * NN + (n0 + col)];
  }
  __syncthreads();
#pragma unroll
  for (int p = 0; p < 4; ++p) {
    int r = row0 + p * 8;
    dst[(long)(n0 + r) * KK + (k0 + col)] = tile[col][r];
  }
}

// ---------------- GEMM + argmin (WMMA f32 16x16x4) ----------------
// block = 256 threads (8 waves). blockIdx.x = M tile (32 rows), blockIdx.y = group.
// Each wave owns interleaved 16-wide N tiles and two 16x16 accumulators (M sub-tiles),
// sharing one B fragment between them.
__global__ void k_argmin(const float* __restrict__ z, const float* __restrict__ emb,
                         const float* __restrict__ e2, int* __restrict__ codes) {
  __shared__ float As[MROWS * LDA];       // A tile, padded stride
  __shared__ float sVal[8][MROWS];
  __shared__ int   sIdx[8][MROWS];

  int g = blockIdx.y;
  int m0 = blockIdx.x * MROWS;
  int tid = threadIdx.x;

  // cooperative load of A tile: 32 rows x 512 k
  for (int i = tid; i < MROWS * KK; i += 256) {
    int r = i >> 9;
    int k = i & (KK - 1);
    int m = m0 + r;
    int b = m >> 8, l = m & (LL - 1);
    As[r * LDA + k] = z[z_index(b, l, g, k)];
  }
  __syncthreads();

  int wave = tid >> 5;
  int lane = tid & 31;
  int half = lane >> 4;
  int lid  = lane & 15;

  float bestVal[2][8];
  int   bestIdx[2][8];
#pragma unroll
  for (int s = 0; s < 2; ++s)
#pragma unroll
    for (int j = 0; j < 8; ++j) { bestVal[s][j] = 3.4e38f; bestIdx[s][j] = 0x7fffffff; }

  const float* embg = emb + (long)g * KN;
  const float* e2g  = e2 + g * NN;

  for (int nt = 0; nt < 8; ++nt) {
    int n0 = nt * 128 + wave * 16;
    v8f accs[2];
    accs[0] = (v8f){};
    accs[1] = (v8f){};
    // accumulate over K in steps of 4; one B frag feeds both M sub-tiles
    for (int kk = 0; kk < KK / 4; ++kk) {
      int krow = kk * 4 + 2 * half;
      v2f bfrag;
      bfrag.x = embg[(long)krow * NN + n0 + lid];
      bfrag.y = embg[(long)(krow + 1) * NN + n0 + lid];
      const float* ap0 = &As[lid * LDA + kk * 4 + 2 * half];
      const float* ap1 = ap0 + 16 * LDA;
      v2f a0; a0.x = ap0[0]; a0.y = ap0[1];
      v2f a1; a1.x = ap1[0]; a1.y = ap1[1];
      accs[0] = __builtin_amdgcn_wmma_f32_16x16x4_f32(
          false, a0, false, bfrag, (short)0, accs[0], false, false);
      accs[1] = __builtin_amdgcn_wmma_f32_16x16x4_f32(
          false, a1, false, bfrag, (short)0, accs[1], false, false);
    }
    // epilogue: dist' = e2[n] - 2*cross ; branchless per-row argmin across the 16
    // lanes of each half (xor masks < 16 stay within the half)
    float e2v = e2g[n0 + lid];
#pragma unroll
    for (int s = 0; s < 2; ++s) {
#pragma unroll
      for (int j = 0; j < 8; ++j) {
        float v = e2v - 2.0f * accs[s][j];
        int   idx = n0 + lid;
#pragma unroll
        for (int mask = 1; mask < 16; mask <<= 1) {
          float ov = __shfl_xor(v, mask, 32);
          int   oi = __shfl_xor(idx, mask, 32);
          int take = (int)(ov < v) | ((int)(ov == v) & (int)(oi < idx));
          v   = take ? ov : v;
          idx = take ? oi : idx;
        }
        int take = (int)(v < bestVal[s][j]) |
                   ((int)(v == bestVal[s][j]) & (int)(idx < bestIdx[s][j]));
        bestVal[s][j] = take ? v   : bestVal[s][j];
        bestIdx[s][j] = take ? idx : bestIdx[s][j];
      }
    }
  }

  // per-wave results -> LDS (row = s*16 + j + 8*half), combine across 8 waves
  if (lid == 0) {
#pragma unroll
    for (int s = 0; s < 2; ++s)
#pragma unroll
      for (int j = 0; j < 8; ++j) {
        sVal[wave][s * 16 + j + 8 * half] = bestVal[s][j];
        sIdx[wave][s * 16 + j + 8 * half] = bestIdx[s][j];
      }
  }
  __syncthreads();
  if (tid < MROWS) {
    float bv = sVal[0][tid]; int bi = sIdx[0][tid];
#pragma unroll
    for (int w = 1; w < 8; ++w) {
      float v = sVal[w][tid]; int i = sIdx[w][tid];
      int take = (int)(v < bv) | ((int)(v == bv) & (int)(i < bi));
      bv = take ? v : bv;
      bi = take ? i : bi;
    }
    codes[(long)g * MM + m0 + tid] = bi;
  }
}

// ---------------- gather q, write xq, loss, counts, scatter acc ----------------
// 8 waves per block; one (g,m) row per wave.
__global__ void k_gather(const float* __restrict__ z, const float* __restrict__ Et,
                         const int* __restrict__ codes, float* __restrict__ out,
                         float* __restrict__ acc, float* __restrict__ counts,
                         float* __restrict__ loss) {
  int wave = threadIdx.x >> 5;
  int lane = threadIdx.x & 31;
  long R = (long)blockIdx.x * 8 + wave;       // 0 .. GR*MM-1
  int g = (int)(R / MM);
  int m = (int)(R - (long)g * MM);
  int code = codes[(long)g * MM + m];
  const float* qrow = Et + ((long)g * NN + code) * KK;
  float* arow = acc + ((long)g * NN + code) * KK;
  int b = m >> 8, l = m & (LL - 1);
  float lsum = 0.0f;
  for (int k = lane; k < KK; k += 32) {
    long zi = z_index(b, l, g, k);
    float qv = qrow[k];
    float zv = z[zi];
    float d = qv - zv;
    lsum += d * d;
    out[zi] = qv;                      // xq (forward value of straight-through)
    atomicAdd(&arow[k], zv);           // emb_sum (transposed)
  }
#pragma unroll
  for (int mask = 1; mask < 32; mask <<= 1) lsum += __shfl_xor(lsum, mask, 32);
  if (lane == 0) {
    atomicAdd(loss, lsum);
    atomicAdd(&counts[g * NN + code], 1.0f);
  }
}

// ---------------- new_cluster_size + per-group sums ----------------
__global__ void k_csum(const float* __restrict__ cluster_size,
                       const float* __restrict__ counts,
                       float* __restrict__ out, float* __restrict__ nsum) {
  __shared__ float red[256];
  int g = blockIdx.x;
  int tid = threadIdx.x;
  float local = 0.0f;
  for (int n = tid; n < NN; n += 256) {
    float ncs = DECAYF * cluster_size[g * NN + n] + OMD * counts[g * NN + n];
    out[OFF_CS + g * NN + n] = ncs;
    local += ncs;
  }
  red[tid] = local;
  __syncthreads();
  for (int s = 128; s > 0; s >>= 1) {
    if (tid < s) red[tid] += red[tid + s];
    __syncthreads();
  }
  if (tid == 0) nsum[g] = red[0];
}

// ---------------- new_embedding + loss finalize ----------------
__global__ void k_final(const float* __restrict__ embedding_mean,
                        const float* __restrict__ acc,
                        const float* __restrict__ nsum,
                        const float* __restrict__ loss,
                        float* __restrict__ out) {
  long t = (long)blockIdx.x * 256 + threadIdx.x;   // 0 .. GR*KN-1
  int g = (int)(t / KN);
  long r = t - (long)g * KN;
  int k = (int)(r / NN);
  int n = (int)(r - (long)k * NN);
  float ncs = out[OFF_CS + g * NN + n];
  float ns = nsum[g];
  float cs = (ncs + EPSF) / (ns + NN * EPSF) * ns;
  float av = acc[((long)g * NN + n) * KK + k];
  float em = embedding_mean[t];
  out[OFF_EMB + t] = (DECAYF * em + OMD * av) / cs;
  if (t == 0) {
    out[OFF_LOSS] = 0.25f * loss[0] / (float)((long)GR * MM * KK);
  }
}

extern "C" void kernel_launch(void* const* d_in, const int* in_sizes, int n_in,
                              void* d_out, int out_size, void* d_ws, size_t ws_size,
                              hipStream_t stream) {
  const float* z            = (const float*)d_in[0];
  const float* embedding    = (const float*)d_in[1];
  const float* cluster_size = (const float*)d_in[2];
  const float* emb_mean     = (const float*)d_in[3];
  float* out = (float*)d_out;
  float* ws  = (float*)d_ws;
  int* codes = (int*)(ws + WS_CODES);

  k_init<<<dim3((WS_ZERO_N + 255) / 256), 256, 0, stream>>>(ws);
  k_e2<<<dim3(GR * NN / 256), 256, 0, stream>>>(embedding, ws + WS_E2);
  k_transpose<<<dim3(KK / 32, NN / 32, GR), 256, 0, stream>>>(embedding, ws + WS_ET);
  k_argmin<<<dim3(MM / MROWS, GR), 256, 0, stream>>>(z, embedding, ws + WS_E2, codes);
  k_gather<<<dim3(GR * MM / 8), 256, 0, stream>>>(z, ws + WS_ET, codes, out,
                                                  ws + WS_ACC, ws + WS_COUNTS, ws + WS_LOSS);
  k_csum<<<dim3(GR), 256, 0, stream>>>(cluster_size, ws + WS_COUNTS, out, ws + WS_NSUM);
  k_final<<<dim3(GR * KN / 256), 256, 0, stream>>>(emb_mean, ws + WS_ACC, ws + WS_NSUM,
                                                   ws + WS_LOSS, out);
}